// FineTuneEsmCNN_78761110274580
// MI455X (gfx1250) — compile-verified
//
#include <hip/hip_runtime.h>
#include <hip/hip_bf16.h>

// Problem constants (from reference)
#define BB   8
#define LL   2048
#define DD   1330
#define H1C  32
#define H2C  16
#define TT   2

#define KCH  64                         // K-chunk staged per TDM transfer (floats)
#define NCH  ((DD + KCH - 1) / KCH)     // 21 chunks; TDM OOB zero-fills past K=1330

typedef float v2f __attribute__((ext_vector_type(2)));
typedef float v8f __attribute__((ext_vector_type(8)));
typedef unsigned int v4u __attribute__((ext_vector_type(4)));
typedef int v4i __attribute__((ext_vector_type(4)));
typedef int v8i __attribute__((ext_vector_type(8)));

#if __has_include(<hip/amd_detail/amd_gfx1250_TDM.h>)
#define TDM_6ARG 1
#endif

// ---------------------------------------------------------------------------
// Issue one 2D TDM tile load: global (tile_w x tile_h, row stride
// row_stride_elems, 4-byte elements) -> LDS (packed row-major).
// tensor_w / tensor_h are the REMAINING tensor extent measured from the tile
// start, so hardware OOB handling zero-fills the ragged K edge in LDS.
// D# field packing per CDNA5 ISA §8 (group0 128b, group1 256b).
// ---------------------------------------------------------------------------
__device__ __forceinline__ void tdm_load_tile_2d(unsigned lds_byte_addr,
                                                 const void* global_tile,
                                                 unsigned tensor_w, unsigned tensor_h,
                                                 unsigned row_stride_elems,
                                                 unsigned tile_w, unsigned tile_h) {
  const unsigned long long ga = (unsigned long long)global_tile;
  const v4u g0 = { 1u,                                        // count=1, user D#
                   lds_byte_addr,                             // lds_addr [63:32]
                   (unsigned)ga,                              // global_addr lo
                   ((unsigned)(ga >> 32) & 0x01FFFFFFu)       // global_addr hi
                     | (2u << 30) };                          // type=2 ("image")
  const v8i g1 = { (int)(2u << 16),                           // data_size=4B; mask=0
                   (int)((tensor_w & 0xFFFFu) << 16),         // tensor_dim0 lo16
                   (int)(((tensor_w >> 16) & 0xFFFFu)
                       | ((tensor_h & 0xFFFFu) << 16)),       // dim0 hi / dim1 lo
                   (int)(((tensor_h >> 16) & 0xFFFFu)
                       | ((tile_w & 0xFFFFu) << 16)),         // dim1 hi / tile_dim0
                   (int)(tile_h & 0xFFFFu),                   // tile_dim1 (tile_dim2=0)
                   (int)row_stride_elems,                     // tensor_dim0_stride lo32
                   0, 0 };                                    // stride hi / dim1_stride
  const v4i gz = { 0, 0, 0, 0 };                              // groups 2/3: dims unused
#if defined(TDM_6ARG)
  const v8i gz8 = { 0, 0, 0, 0, 0, 0, 0, 0 };
  __builtin_amdgcn_tensor_load_to_lds(g0, g1, gz, gz, gz8, 0);
#else
  __builtin_amdgcn_tensor_load_to_lds(g0, g1, gz, gz, 0);
#endif
}

// ---------------------------------------------------------------------------
// Kernel 1: h[BL,32] = x[BL,1330] @ w1[1330,32] + b1
// One wave32 per 16-row M-tile. Double-buffered TDM pipeline:
//   TENSOR_LOAD_TO_LDS stages the 16xKCH x-tile and KCHx32 w1-tile into LDS
//   while WMMA consumes the previous buffers; s_wait_tensorcnt(2) overlaps
//   exactly one chunk of DMA with compute. TDM OOB zero-fill past K=1330
//   makes the inner loop guard-free.
// WMMA f32 16x16x4 layouts:
//   A: lane<16 holds row M=lane, K={k,k+1}; lane+16 same row, K={k+2,k+3}
//   B: VGPR0 = K row k (lanes 0-15) / k+2 (16-31), VGPR1 = k+1 / k+3, N=lane%16
//   C/D: VGPR v -> (M = v + 8*half, N = lane%16)
// ---------------------------------------------------------------------------
__global__ __launch_bounds__(32)
void emis_gemm_wmma_tdm(const float* __restrict__ x,
                        const float* __restrict__ w1,
                        const float* __restrict__ bias,
                        float* __restrict__ hout) {
  __shared__ __align__(16) float sx[2][16 * KCH];   // 2 x 4 KB  (x tiles)
  __shared__ __align__(16) float sw[2][KCH * H1C];  // 2 x 8 KB  (w1 tiles)

  const int lane = threadIdx.x & 31;
  const int row0 = blockIdx.x * 16;                 // 1024 blocks, 16 rows each
  const int half = lane >> 4;
  const int r    = lane & 15;

  const unsigned sxa0 = (unsigned)(unsigned long long)(void*)&sx[0][0];
  const unsigned sxa1 = (unsigned)(unsigned long long)(void*)&sx[1][0];
  const unsigned swa0 = (unsigned)(unsigned long long)(void*)&sw[0][0];
  const unsigned swa1 = (unsigned)(unsigned long long)(void*)&sw[1][0];

#define ISSUE_CHUNK(cc, bb_)                                                   \
  do {                                                                         \
    const unsigned k0_ = (unsigned)(cc) * KCH;                                 \
    tdm_load_tile_2d((bb_) ? sxa1 : sxa0, x + (size_t)row0 * DD + k0_,         \
                     (unsigned)DD - k0_, 16u, (unsigned)DD, KCH, 16u);         \
    tdm_load_tile_2d((bb_) ? swa1 : swa0, w1 + (size_t)k0_ * H1C,              \
                     (unsigned)H1C, (unsigned)DD - k0_, (unsigned)H1C,         \
                     (unsigned)H1C, KCH);                                      \
  } while (0)

  ISSUE_CHUNK(0, 0);

  v8f c0 = {};   // N = 0..15
  v8f c1 = {};   // N = 16..31

  for (int c = 0; c < NCH; ++c) {
    const int buf = c & 1;
    if (c + 1 < NCH) {
      ISSUE_CHUNK(c + 1, buf ^ 1);
      __builtin_amdgcn_s_wait_tensorcnt((short)2);  // chunk c's 2 TDM ops done
    } else {
      __builtin_amdgcn_s_wait_tensorcnt((short)0);
    }
    asm volatile("" ::: "memory");                  // fence TDM LDS writes vs ds reads

    const float* __restrict__ ax = &sx[buf][0];
    const float* __restrict__ bw = &sw[buf][0];
#pragma unroll
    for (int t = 0; t < KCH / 4; ++t) {
      const int kl = t * 4 + 2 * half;
      const v2f a = *(const v2f*)(ax + r * KCH + kl);   // ds_load_b64
      v2f b0, b1;
      b0.x = bw[kl * H1C + r];
      b0.y = bw[(kl + 1) * H1C + r];
      b1.x = bw[kl * H1C + 16 + r];
      b1.y = bw[(kl + 1) * H1C + 16 + r];
      c0 = __builtin_amdgcn_wmma_f32_16x16x4_f32(false, a, false, b0,
                                                 (short)0, c0, false, false);
      c1 = __builtin_amdgcn_wmma_f32_16x16x4_f32(false, a, false, b1,
                                                 (short)0, c1, false, false);
    }
    asm volatile("" ::: "memory");                  // keep reads inside this stage
  }
#undef ISSUE_CHUNK

  const float bv0 = bias[r];
  const float bv1 = bias[16 + r];
#pragma unroll
  for (int v = 0; v < 8; ++v) {
    const int m = row0 + v + 8 * half;
    hout[(size_t)m * H1C + r]      = c0[v] + bv0;
    hout[(size_t)m * H1C + 16 + r] = c1[v] + bv1;
  }
}

// ---------------------------------------------------------------------------
// Kernel 2: em[BL,2] = relu(conv1d(h, cw, pad=1) + cb) @ w2 + b2
// One thread per (b,l); weights staged in LDS.
// ---------------------------------------------------------------------------
__global__ __launch_bounds__(256)
void conv_head(const float* __restrict__ h,
               const float* __restrict__ cw,   // [16][32][3]
               const float* __restrict__ cb,   // [16]
               const float* __restrict__ w2,   // [16][2]
               const float* __restrict__ b2,   // [2]
               float* __restrict__ em) {
  __shared__ float s_cw[H2C * H1C * 3];
  __shared__ float s_cb[H2C];
  __shared__ float s_w2[H2C * TT];
  __shared__ float s_b2[TT];

  for (int i = threadIdx.x; i < H2C * H1C * 3; i += 256) s_cw[i] = cw[i];
  if (threadIdx.x < H2C)      s_cb[threadIdx.x] = cb[threadIdx.x];
  if (threadIdx.x < H2C * TT) s_w2[threadIdx.x] = w2[threadIdx.x];
  if (threadIdx.x < TT)       s_b2[threadIdx.x] = b2[threadIdx.x];
  __syncthreads();

  const int idx = blockIdx.x * 256 + threadIdx.x;   // b*LL + l
  if (idx >= BB * LL) return;
  const int l = idx & (LL - 1);

  float acc[H2C];
#pragma unroll
  for (int c2 = 0; c2 < H2C; ++c2) acc[c2] = s_cb[c2];

#pragma unroll
  for (int k = 0; k < 3; ++k) {
    const int ll = l + k - 1;
    if (ll < 0 || ll >= LL) continue;               // zero padding
    const float* __restrict__ hr = h + (size_t)(idx + k - 1) * H1C;
    for (int c1 = 0; c1 < H1C; ++c1) {
      const float hv = hr[c1];
#pragma unroll
      for (int c2 = 0; c2 < H2C; ++c2)
        acc[c2] = fmaf(hv, s_cw[(c2 * H1C + c1) * 3 + k], acc[c2]);
    }
  }

  float e0 = s_b2[0], e1 = s_b2[1];
#pragma unroll
  for (int c2 = 0; c2 < H2C; ++c2) {
    const float a = fmaxf(acc[c2], 0.f);
    e0 = fmaf(a, s_w2[c2 * TT + 0], e0);
    e1 = fmaf(a, s_w2[c2 * TT + 1], e1);
  }
  em[(size_t)idx * TT + 0] = e0;
  em[(size_t)idx * TT + 1] = e1;
}

// ---------------------------------------------------------------------------
// Kernel 3: CRF per batch element. One block per b (grid=8, block=64).
//   lane 0  : gold path score + forward logsumexp recursion -> llh[b]
//   lane 32 : Viterbi forward (history in LDS) + backtrack -> tags
// Tags for padded positions are zero-filled by the whole block first.
// ---------------------------------------------------------------------------
__global__ __launch_bounds__(64)
void crf_kernel(const float* __restrict__ em,
                const int* __restrict__ tokens_length,
                const int* __restrict__ labels,
                const float* __restrict__ start_trans,
                const float* __restrict__ end_trans,
                const float* __restrict__ trans,
                float* __restrict__ llh,
                float* __restrict__ tags_out) {
  __shared__ unsigned char hist[LL][TT];            // 4 KB of the 320 KB LDS

  const int b   = blockIdx.x;
  const int len = tokens_length[b];
  const float* __restrict__ emb = em + (size_t)b * LL * TT;
  const int* __restrict__ lab   = labels + (size_t)b * LL;
  float* __restrict__ tb        = tags_out + (size_t)b * LL;

  // zero-fill tags (masked region must be 0; d_out is poisoned by harness)
  for (int l = threadIdx.x; l < LL; l += 64) tb[l] = 0.f;
  __syncthreads();

  const float t00 = trans[0], t01 = trans[1], t10 = trans[2], t11 = trans[3];

  if (threadIdx.x == 0) {
    // ---- gold score + forward algorithm ----
    int prev = lab[0];
    float score = start_trans[prev] + emb[prev];
    float a0 = start_trans[0] + emb[0];
    float a1 = start_trans[1] + emb[1];
    for (int i = 1; i < len; ++i) {
      const int   li = lab[i];
      const float e0 = emb[i * TT + 0];
      const float e1 = emb[i * TT + 1];
      score += trans[prev * TT + li] + (li ? e1 : e0);
      prev = li;

      const float u0 = a0 + t00, v0 = a1 + t10;
      const float m0 = fmaxf(u0, v0);
      const float n0 = m0 + logf(expf(u0 - m0) + expf(v0 - m0)) + e0;
      const float u1 = a0 + t01, v1 = a1 + t11;
      const float m1 = fmaxf(u1, v1);
      const float n1 = m1 + logf(expf(u1 - m1) + expf(v1 - m1)) + e1;
      a0 = n0; a1 = n1;
    }
    score += end_trans[lab[len - 1]];
    const float f0 = a0 + end_trans[0];
    const float f1 = a1 + end_trans[1];
    const float mm = fmaxf(f0, f1);
    const float norm = mm + logf(expf(f0 - mm) + expf(f1 - mm));
    llh[b] = score - norm;
  } else if (threadIdx.x == 32) {
    // ---- Viterbi ----
    float v0 = start_trans[0] + emb[0];
    float v1 = start_trans[1] + emb[1];
    for (int i = 1; i < len; ++i) {
      const float e0 = emb[i * TT + 0];
      const float e1 = emb[i * TT + 1];
      const float s00 = v0 + t00, s10 = v1 + t10;   // -> cur = 0
      const float s01 = v0 + t01, s11 = v1 + t11;   // -> cur = 1
      hist[i][0] = (unsigned char)((s10 > s00) ? 1 : 0);  // argmax tie -> 0
      hist[i][1] = (unsigned char)((s11 > s01) ? 1 : 0);
      const float n0 = fmaxf(s00, s10) + e0;
      const float n1 = fmaxf(s01, s11) + e1;
      v0 = n0; v1 = n1;
    }
    int tag = ((v1 + end_trans[1]) > (v0 + end_trans[0])) ? 1 : 0;
    tb[len - 1] = (float)tag;
    for (int i = len - 1; i >= 1; --i) {
      tag = hist[i][tag];
      tb[i - 1] = (float)tag;
    }
  }
}

// ---------------------------------------------------------------------------
// Kernel 4: neg_loss = -sum_b llh[b]  (single-thread, deterministic order)
// ---------------------------------------------------------------------------
__global__ void finalize_loss(const float* __restrict__ llh,
                              float* __restrict__ out) {
  if (threadIdx.x == 0 && blockIdx.x == 0) {
    float s = 0.f;
#pragma unroll
    for (int b = 0; b < BB; ++b) s += llh[b];
    out[0] = -s;
  }
}

// ---------------------------------------------------------------------------
extern "C" void kernel_launch(void* const* d_in, const int* in_sizes, int n_in,
                              void* d_out, int out_size, void* d_ws, size_t ws_size,
                              hipStream_t stream) {
  (void)in_sizes; (void)n_in; (void)out_size; (void)ws_size;

  const float* x            = (const float*)d_in[0];
  const int*   tokens_len   = (const int*)  d_in[1];
  const int*   labels       = (const int*)  d_in[2];
  const float* w1           = (const float*)d_in[3];
  const float* b1           = (const float*)d_in[4];
  const float* conv_w       = (const float*)d_in[5];
  const float* conv_b       = (const float*)d_in[6];
  const float* w2           = (const float*)d_in[7];
  const float* b2           = (const float*)d_in[8];
  const float* start_trans  = (const float*)d_in[9];
  const float* end_trans    = (const float*)d_in[10];
  const float* trans        = (const float*)d_in[11];

  float* out = (float*)d_out;                 // out[0] = neg_loss, out[1..] = tags

  float* h   = (float*)d_ws;                  // [BB*LL, 32]  = 2 MB
  float* em  = h  + (size_t)BB * LL * H1C;    // [BB*LL, 2]   = 128 KB
  float* llh = em + (size_t)BB * LL * TT;     // [BB]

  // one wave32 workgroup per 16-row M-tile (1024 tiles)
  emis_gemm_wmma_tdm<<<BB * LL / 16, 32, 0, stream>>>(x, w1, b1, h);
  conv_head<<<(BB * LL + 255) / 256, 256, 0, stream>>>(h, conv_w, conv_b, w2, b2, em);
  crf_kernel<<<BB, 64, 0, stream>>>(em, tokens_len, labels, start_trans,
                                    end_trans, trans, llh, out + 1);
  finalize_loss<<<1, 1, 0, stream>>>(llh, out);
}